// VQ_VAE_65987877536129
// MI455X (gfx1250) — compile-verified
//
#include <hip/hip_runtime.h>
#include <hip/hip_bf16.h>

#define N_ROWS 16384
#define DIM 512
#define K_CODES 1024
// TDM-padded LDS row stride: 512 dwords + 2x6 pad dwords = 524 floats/row.
// Row r: cols [0,256) at r*524 + c ; cols [256,512) at r*524 + 262 + (c-256).
#define LDE 524
#define EHALF 262   // K-half 1 base offset within a row

typedef __attribute__((ext_vector_type(2))) float v2f;
typedef __attribute__((ext_vector_type(8))) float v8f;
typedef __attribute__((ext_vector_type(4))) unsigned int u32x4;
typedef __attribute__((ext_vector_type(4))) int i32x4;
typedef __attribute__((ext_vector_type(8))) int i32x8;

// ws layout (floats): [0] loss accumulator, [16..16+1023] enorm, then int idx[16384]

__global__ void vq_enorm(const float* __restrict__ E, float* __restrict__ wsf) {
    int k = blockIdx.x * blockDim.x + threadIdx.x;
    if (k == 0) wsf[0] = 0.0f;                 // zero loss accumulator each call
    if (k < K_CODES) {
        const float4* p = (const float4*)(E + (long)k * DIM);
        float s = 0.0f;
        for (int i = 0; i < DIM / 4; ++i) {
            float4 v = p[i];
            s += v.x * v.x + v.y * v.y + v.z * v.z + v.w * v.w;
        }
        wsf[16 + k] = s;
    }
}

__global__ __launch_bounds__(256)
void vq_dist_argmin(const float* __restrict__ X, const float* __restrict__ E,
                    const float* __restrict__ enorm, int* __restrict__ wsIdx,
                    float* __restrict__ outIdx)
{
    __shared__ float Elds[2][16 * LDE];    // 2 x 33,536 B (double-buffered E tile)
    __shared__ float scratch[4 * 32 * 8];  // 4,096 B (K-half partial exchange)

    const int tid    = threadIdx.x;
    const int wave   = tid >> 5;
    const int lane   = tid & 31;
    const int l16    = lane & 15;
    const int kh     = wave >> 2;            // K-half: 0 => depth [0,256), 1 => [256,512)
    const int rt     = wave & 3;             // row tile 0..3 (16 rows each)
    const int khalf2 = (lane >> 4) << 1;     // 0 or 2 (A/B lane K sub-offset)
    const int rowBase = blockIdx.x * 64 + rt * 16;
    const int gRow    = rowBase + l16;

    // A fragments for this wave's 16 rows x 256 depths, held in registers.
    // Fold -2 scale here: score = ||e||^2 + (-2x)·e
    v2f a[64];
    const int kbase = kh * 256 + khalf2;
    #pragma unroll
    for (int s = 0; s < 64; ++s) {
        const float* p = X + (long)gRow * DIM + kbase + 4 * s;
        v2f av; av[0] = -2.0f * p[0]; av[1] = -2.0f * p[1];
        a[s] = av;
    }

    float best[8];
    int   bidx[8];
    #pragma unroll
    for (int j = 0; j < 8; ++j) { best[j] = 3.4e38f; bidx[j] = 0; }

#if __has_builtin(__builtin_amdgcn_tensor_load_to_lds) && __has_builtin(__builtin_amdgcn_s_wait_tensorcnt)
    // ---- Tensor Data Mover staging (async DMA global -> LDS, double-buffered) ----
    auto issue_tdm = [&](int tile, int buf) {
        unsigned lds_base = (unsigned)(unsigned long long)(uintptr_t)&Elds[buf][0];
        unsigned long long ga =
            (unsigned long long)(uintptr_t)(E + (unsigned long long)tile * 16u * DIM);
        u32x4 g0;
        g0[0] = 1u;                                   // count=1, is_restore=0, gather off
        g0[1] = lds_base;                             // lds_addr (bytes)
        g0[2] = (unsigned)(ga & 0xFFFFFFFFull);       // global_addr[31:0]
        g0[3] = (unsigned)((ga >> 32) & 0x01FFFFFFull) | 0x80000000u; // addr[56:32] | type=2
        i32x8 g1;
        // data_size=4B(2), pad_enable, pad_interval=7 (256 dw), pad_amount=5 (6 dw)
        g1[0] = (int)((2u << 16) | (1u << 20) | (7u << 22) | (5u << 25));
        g1[1] = (int)(512u << 16);   // tensor_dim0 = 512 (low16 at bits 63:48)
        g1[2] = (int)(16u << 16);    // tensor_dim0 hi16=0 | tensor_dim1 = 16
        g1[3] = (int)(512u << 16);   // tensor_dim1 hi16=0 | tile_dim0 = 512
        g1[4] = 16;                  // tile_dim1 = 16, tile_dim2 = 0
        g1[5] = 512;                 // tensor_dim0_stride = 512 (elements)
        g1[6] = 0;                   // stride hi | tensor_dim1_stride lo
        g1[7] = 0;
        i32x4 z4 = {0, 0, 0, 0};             // groups 2/3 unused for 2D tiles
        i32x8 z8 = {0, 0, 0, 0, 0, 0, 0, 0}; // extra group (6-arg toolchain form)
        __builtin_amdgcn_tensor_load_to_lds(g0, g1, z4, z4, z8, 0);
    };

    if (wave == 0) issue_tdm(0, 0);

    for (int t = 0; t < 64; ++t) {
        const int buf = t & 1;
        if (wave == 0) {
            if (t + 1 < 64) {
                issue_tdm(t + 1, buf ^ 1);              // overlap next tile's DMA
                __builtin_amdgcn_s_wait_tensorcnt(1);   // in-order => tile t resident
            } else {
                __builtin_amdgcn_s_wait_tensorcnt(0);
            }
        }
        __syncthreads();   // E tile t visible to all waves
#else
    // ---- Fallback: synchronous cooperative staging into the same layout ----
    const float2* E2 = (const float2*)E;
    for (int t = 0; t < 64; ++t) {
        const int buf = t & 1;
        __syncthreads();
        #pragma unroll
        for (int j = 0; j < 16; ++j) {
            int f  = tid + 256 * j;          // float2 index 0..4095 (16 rows x 256)
            int r  = f >> 8;                 // row
            int c2 = f & 255;                // float2 col
            float2 v = E2[t * 4096 + f];
            int off = r * LDE + 4 * (c2 >> 1) + 2 * (c2 & 1);
            if (c2 >= 128) off += 6;         // TDM-style mid-row pad
            *(float2*)&Elds[buf][off] = v;
        }
        __syncthreads();
#endif

        v8f acc;
        {
            float s0 = kh ? enorm[t * 16 + l16] : 0.0f;  // seed C with ||e||^2 once
            #pragma unroll
            for (int j = 0; j < 8; ++j) acc[j] = s0;
        }

        const float* eb = &Elds[buf][l16 * LDE + kh * EHALF + khalf2];
        #pragma unroll
        for (int s = 0; s < 64; ++s) {
            v2f b = *(const v2f*)(eb + 4 * s);
            acc = __builtin_amdgcn_wmma_f32_16x16x4_f32(
                      false, a[s], false, b, (short)0, acc, false, false);
        }

        if (kh == 0) {
            float* sp = &scratch[(rt * 32 + lane) * 8];
            #pragma unroll
            for (int j = 0; j < 8; ++j) sp[j] = acc[j];
        }
        __syncthreads();   // partials visible; all reads of buf complete
        if (kh == 1) {
            const float* sp = &scratch[(rt * 32 + lane) * 8];
            int code = t * 16 + l16;
            #pragma unroll
            for (int j = 0; j < 8; ++j) {
                float sc = acc[j] + sp[j];
                if (sc < best[j]) { best[j] = sc; bidx[j] = code; }
            }
        }
    }

    if (kh == 1) {
        // Butterfly min+argmin across the 16 lanes holding the same rows.
        #pragma unroll
        for (int m = 1; m < 16; m <<= 1) {
            #pragma unroll
            for (int j = 0; j < 8; ++j) {
                float ov = __shfl_xor(best[j], m, 32);
                int   oi = __shfl_xor(bidx[j], m, 32);
                if (ov < best[j] || (ov == best[j] && oi < bidx[j])) {
                    best[j] = ov; bidx[j] = oi;
                }
            }
        }
        if (l16 == 0) {
            // lane 0 holds rows rowBase+0..7, lane 16 holds rows rowBase+8..15
            int rbase = rowBase + ((lane >> 4) << 3);
            #pragma unroll
            for (int j = 0; j < 8; ++j) {
                wsIdx[rbase + j]  = bidx[j];
                outIdx[rbase + j] = (float)bidx[j];
            }
        }
    }
}

__global__ __launch_bounds__(128)
void vq_gather_loss(const float* __restrict__ X, const float* __restrict__ E,
                    const int* __restrict__ wsIdx, float* __restrict__ outQ,
                    float* __restrict__ lossAcc)
{
    __shared__ float wpart[4];
    const int n = blockIdx.x;
    const int t = threadIdx.x;
    const int idx = wsIdx[n];

    const float4 q = ((const float4*)(E + (long)idx * DIM))[t];
    const float4 x = ((const float4*)(X + (long)n   * DIM))[t];

    // outQ base (d_out+1) is only 4B-aligned -> scalar stores
    float* o = outQ + (long)n * DIM + t * 4;
    o[0] = q.x; o[1] = q.y; o[2] = q.z; o[3] = q.w;

    float dx = q.x - x.x, dy = q.y - x.y, dz = q.z - x.z, dw = q.w - x.w;
    float s = dx * dx + dy * dy + dz * dz + dw * dw;
    #pragma unroll
    for (int m = 16; m > 0; m >>= 1) s += __shfl_down(s, m, 32);
    if ((t & 31) == 0) wpart[t >> 5] = s;
    __syncthreads();
    if (t == 0) {
        float tot = wpart[0] + wpart[1] + wpart[2] + wpart[3];
        atomicAdd(lossAcc, tot);
    }
}

__global__ void vq_finalize(const float* __restrict__ lossAcc, float* __restrict__ out0) {
    // loss = q_latent + 0.25*e_latent = 1.25 * mean((q - x)^2)
    *out0 = 1.25f * (*lossAcc) * (1.0f / 8388608.0f);   // N_ROWS*DIM
}

extern "C" void kernel_launch(void* const* d_in, const int* in_sizes, int n_in,
                              void* d_out, int out_size, void* d_ws, size_t ws_size,
                              hipStream_t stream) {
    (void)in_sizes; (void)n_in; (void)out_size; (void)ws_size;

    const float* X = (const float*)d_in[0];   // inputs [64,256,512] f32
    const float* E = (const float*)d_in[1];   // emb_weight [1024,512] f32

    float* out    = (float*)d_out;
    float* wsf    = (float*)d_ws;
    float* enorm  = wsf + 16;
    int*   wsIdx  = (int*)(wsf + 16 + K_CODES);
    float* outQ   = out + 1;                      // quantized [16384*512]
    float* outIdx = out + 1 + (long)N_ROWS * DIM; // indices as float [16384]

    vq_enorm      <<<(K_CODES + 255) / 256, 256, 0, stream>>>(E, wsf);
    vq_dist_argmin<<<N_ROWS / 64,           256, 0, stream>>>(X, E, enorm, wsIdx, outIdx);
    vq_gather_loss<<<N_ROWS,                128, 0, stream>>>(X, E, wsIdx, outQ, wsf);
    vq_finalize   <<<1,                       1, 0, stream>>>(wsf, out);
}